// Attn_39625368273382
// MI455X (gfx1250) — compile-verified
//
#include <hip/hip_runtime.h>

typedef __attribute__((ext_vector_type(2))) float v2f;
typedef __attribute__((ext_vector_type(4))) float v4f;
typedef __attribute__((ext_vector_type(8))) float v8f;

#define HID 1024
#define SEQ 32768

// ---------------------------------------------------------------------------
// K1: u[h] = sum_g W[g][h] * hidden[g]   (W is [HID][HID] row-major)
// grid: 16 blocks x 256 threads; each block covers 64 columns h,
// 4 g-chunks of 256 reduced through LDS.
// ---------------------------------------------------------------------------
__global__ __launch_bounds__(256) void k1_proj_vec(const float* __restrict__ W,
                                                   const float* __restrict__ hidden,
                                                   float* __restrict__ u) {
    __shared__ float red[4][64];
    const int tx = threadIdx.x & 63;   // column within block's 64-col slab
    const int ty = threadIdx.x >> 6;   // g-chunk 0..3
    const int h  = blockIdx.x * 64 + tx;
    const int g0 = ty * 256;
    float acc = 0.0f;
#pragma unroll 8
    for (int g = g0; g < g0 + 256; ++g)
        acc += W[(size_t)g * HID + h] * hidden[g];
    red[ty][tx] = acc;
    __syncthreads();
    if (ty == 0)
        u[h] = red[0][tx] + red[1][tx] + red[2][tx] + red[3][tx];
}

// ---------------------------------------------------------------------------
// K2: energies[s] = enc[s,:] . u  via V_WMMA_F32_16X16X4_F32.
// One wave handles a 16-row tile. A = enc tile (ISA 16x4 f32 layout:
// lanes 0-15 -> K=0,1 ; lanes 16-31 -> K=2,3). B = u chunk replicated across
// all 16 N columns, so D[m][n] = energy[m] for every n.
// K is permuted so each lane reads a contiguous float4 covering two WMMA steps.
// grid: 256 blocks x 256 threads (8 waves/block -> 2048 tiles).
// ---------------------------------------------------------------------------
__global__ __launch_bounds__(256) void k2_energies(const float* __restrict__ enc,
                                                   const float* __restrict__ u,
                                                   float* __restrict__ energies) {
    __shared__ float su[HID];
#pragma unroll
    for (int i = threadIdx.x; i < HID; i += 256) su[i] = u[i];   // uniform trips
    __syncthreads();

    const int wave = threadIdx.x >> 5;
    const int lane = threadIdx.x & 31;
    const int tile = blockIdx.x * 8 + wave;
    const int rb   = tile * 16;
    const int m    = lane & 15;            // row within tile (A-matrix M)
    const int koff = (lane >> 4) << 2;     // float offset: 0 (K-pair 0,1) or 4 (K-pair 2,3)

    const float* rowp = enc + (size_t)(rb + m) * HID + koff;
    const float* up   = su + koff;

    v8f acc = {0.f, 0.f, 0.f, 0.f, 0.f, 0.f, 0.f, 0.f};

#pragma unroll 4
    for (int g = 0; g < 128; ++g) {
        v4f a4 = *(const v4f*)(rowp + g * 8);   // global_load_b128
        v4f u4 = *(const v4f*)(up   + g * 8);   // ds_load_b128
        v2f a0 = {a4.x, a4.y}, b0 = {u4.x, u4.y};
        v2f a1 = {a4.z, a4.w}, b1 = {u4.z, u4.w};
        acc = __builtin_amdgcn_wmma_f32_16x16x4_f32(false, a0, false, b0,
                                                    (short)0, acc, false, false);
        acc = __builtin_amdgcn_wmma_f32_16x16x4_f32(false, a1, false, b1,
                                                    (short)0, acc, false, false);
    }

    // D layout: VGPR r -> row r (lanes 0-15) / row r+8 (lanes 16-31); all N equal.
    if (lane == 0) {
#pragma unroll
        for (int r = 0; r < 8; ++r) energies[rb + r] = acc[r];
    } else if (lane == 16) {
#pragma unroll
        for (int r = 0; r < 8; ++r) energies[rb + 8 + r] = acc[r];
    }
}

// ---------------------------------------------------------------------------
// K3a: per-block (max, sum exp) partials. 64 blocks x 256 threads, 512 elems each.
// ---------------------------------------------------------------------------
__global__ __launch_bounds__(256) void k3a_partials(const float* __restrict__ e,
                                                    float* __restrict__ pm,
                                                    float* __restrict__ ps) {
    __shared__ float sm[256];
    __shared__ float ss[256];
    const int t  = threadIdx.x;
    const int i0 = blockIdx.x * 512 + t;
    const float x0 = e[i0];
    const float x1 = e[i0 + 256];
    sm[t] = fmaxf(x0, x1);
    __syncthreads();
    for (int s = 128; s > 0; s >>= 1) {
        if (t < s) sm[t] = fmaxf(sm[t], sm[t + s]);
        __syncthreads();
    }
    const float bm = sm[0];
    ss[t] = __expf(x0 - bm) + __expf(x1 - bm);
    __syncthreads();
    for (int s = 128; s > 0; s >>= 1) {
        if (t < s) ss[t] += ss[t + s];
        __syncthreads();
    }
    if (t == 0) { pm[blockIdx.x] = bm; ps[blockIdx.x] = ss[0]; }
}

// ---------------------------------------------------------------------------
// K3b: combine 64 partials -> global (M, S). 1 block x 64 threads.
// ---------------------------------------------------------------------------
__global__ __launch_bounds__(64) void k3b_combine(const float* __restrict__ pm,
                                                  const float* __restrict__ ps,
                                                  float* __restrict__ g) {
    __shared__ float sm[64];
    __shared__ float ss[64];
    const int t = threadIdx.x;
    sm[t] = pm[t];
    __syncthreads();
    for (int s = 32; s > 0; s >>= 1) {
        if (t < s) sm[t] = fmaxf(sm[t], sm[t + s]);
        __syncthreads();
    }
    const float M = sm[0];
    __syncthreads();
    ss[t] = ps[t] * __expf(pm[t] - M);
    __syncthreads();
    for (int s = 32; s > 0; s >>= 1) {
        if (t < s) ss[t] += ss[t + s];
        __syncthreads();
    }
    if (t == 0) { g[0] = M; g[1] = ss[0]; }
}

// ---------------------------------------------------------------------------
// K3c: out[i] = exp(e[i]-M)/S. 128 blocks x 256 threads.
// ---------------------------------------------------------------------------
__global__ __launch_bounds__(256) void k3c_final(const float* __restrict__ e,
                                                 const float* __restrict__ g,
                                                 float* __restrict__ out) {
    const int i = blockIdx.x * 256 + threadIdx.x;
    const float M = g[0];
    const float S = g[1];
    out[i] = __expf(e[i] - M) / S;
}

extern "C" void kernel_launch(void* const* d_in, const int* in_sizes, int n_in,
                              void* d_out, int out_size, void* d_ws, size_t ws_size,
                              hipStream_t stream) {
    // setup_inputs order: hidden[1024], encoder_outputs[1,32768,1024],
    //                     attn_w[1024,1024], attn_b[1024] (bias cancels in softmax)
    const float* hidden = (const float*)d_in[0];
    const float* enc    = (const float*)d_in[1];
    const float* W      = (const float*)d_in[2];
    float* out = (float*)d_out;

    float* ws  = (float*)d_ws;
    float* u   = ws;                 // 1024
    float* eng = ws + HID;           // 32768
    float* pm  = eng + SEQ;          // 64
    float* ps  = pm + 64;            // 64
    float* gg  = ps + 64;            // 2

    k1_proj_vec<<<16, 256, 0, stream>>>(W, hidden, u);
    k2_energies<<<256, 256, 0, stream>>>(enc, u, eng);
    k3a_partials<<<64, 256, 0, stream>>>(eng, pm, ps);
    k3b_combine<<<1, 64, 0, stream>>>(pm, ps, gg);
    k3c_final<<<SEQ / 256, 256, 0, stream>>>(eng, gg, out);
}